// GaussianGraphNetwork_88098369175852
// MI455X (gfx1250) — compile-verified
//
#include <hip/hip_runtime.h>
#include <math.h>

#define NODES 50000
#define EDGES 800000

typedef __attribute__((ext_vector_type(2))) float v2f;
typedef __attribute__((ext_vector_type(8))) float v8f;

// ---------------------------------------------------------------------------
// Monotone float <-> uint mapping so float max can use hardware u32 atomic max
// ---------------------------------------------------------------------------
__device__ __forceinline__ unsigned encf(float f) {
  unsigned b = __float_as_uint(f);
  return (b & 0x80000000u) ? ~b : (b | 0x80000000u);
}
__device__ __forceinline__ float decf(unsigned e) {
  return __uint_as_float((e & 0x80000000u) ? (e ^ 0x80000000u) : ~e);
}
#define ENC_NEG_INF 0x007FFFFFu  // encf(-inf)

// ---------------------------------------------------------------------------
// C[M x 64] = A[M x KDIM] @ B[KDIM x 64] (+ bias), M multiple of 16.
// One wave computes a 16-row strip (four 16x16 tiles) via V_WMMA_F32_16X16X4_F32.
// ---------------------------------------------------------------------------
template <int KDIM>
__global__ void __launch_bounds__(256)
gemm64_wmma_kernel(const float* __restrict__ A, const float* __restrict__ B,
                   const float* __restrict__ bias, float* __restrict__ C, int M) {
  const int wave = (blockIdx.x * blockDim.x + threadIdx.x) >> 5;
  const int lane = threadIdx.x & 31;
  const int m0 = wave * 16;
  if (m0 >= M) return;  // whole-wave uniform exit: EXEC stays all-ones for WMMA

  const int halfk = (lane >> 4) * 2;   // lanes 0-15 hold K {4t,4t+1}; 16-31 hold {4t+2,4t+3}
  const int mrow  = m0 + (lane & 15);
  const int ncol  = lane & 15;

  // A fragments for all K steps (KDIM/4 x v2f), kept in registers.
  v2f afrag[KDIM / 4];
#pragma unroll
  for (int t = 0; t < KDIM / 4; ++t) {
    const int kb = 4 * t + halfk;
    afrag[t].x = A[(size_t)mrow * KDIM + kb + 0];
    afrag[t].y = A[(size_t)mrow * KDIM + kb + 1];
  }

#pragma unroll
  for (int n = 0; n < 4; ++n) {
    const int col = n * 16 + ncol;
    v8f acc = {0.f, 0.f, 0.f, 0.f, 0.f, 0.f, 0.f, 0.f};
#pragma unroll
    for (int t = 0; t < KDIM / 4; ++t) {
      const int kr = 4 * t + halfk;
      v2f b;
      b.x = B[(size_t)(kr + 0) * 64 + col];
      b.y = B[(size_t)(kr + 1) * 64 + col];
      acc = __builtin_amdgcn_wmma_f32_16x16x4_f32(
          /*neg_a=*/false, afrag[t], /*neg_b=*/false, b,
          /*c_mod=*/(short)0, acc, /*reuse_a=*/false, /*reuse_b=*/false);
    }
    const float bb = bias ? bias[col] : 0.f;
#pragma unroll
    for (int r = 0; r < 8; ++r) {
      const int row = m0 + r + 8 * (lane >> 4);   // C/D layout: vgpr r -> M=r (+8 for hi half)
      C[(size_t)row * 64 + col] = acc[r] + bb;
    }
  }
}

// ---------------------------------------------------------------------------
// Init per-layer softmax state: Menc = enc(-inf), Ssum = 0, Acc = 0
// ---------------------------------------------------------------------------
__global__ void init_state_kernel(unsigned* __restrict__ Menc, float* __restrict__ Ssum,
                                  float* __restrict__ Acc, int n, int H) {
  const int i = blockIdx.x * blockDim.x + threadIdx.x;
  if (i < n * 64) Acc[i] = 0.f;
  if (i < n * H) { Ssum[i] = 0.f; Menc[i] = ENC_NEG_INF; }
}

// ---------------------------------------------------------------------------
// Pass 1: alpha[e,h] = <q[dst], k[src]+e> / sqrt(C); scatter-max into Menc.
// Wave-per-edge: lane handles channels {lane, lane+32}. H==2 -> C=32 halves.
// ---------------------------------------------------------------------------
__global__ void __launch_bounds__(256)
edge_alpha_kernel(const float* __restrict__ Q, const float* __restrict__ Kf,
                  const float* __restrict__ Ee, const int* __restrict__ ei,
                  float* __restrict__ alpha, unsigned* __restrict__ Menc,
                  int H, float rscale) {
  const int e = (blockIdx.x * blockDim.x + threadIdx.x) >> 5;
  if (e >= EDGES) return;
  const int lane = threadIdx.x & 31;
  const int src = ei[e], dst = ei[EDGES + e];

  const float* qr = Q + (size_t)dst * 64;
  const float* kr = Kf + (size_t)src * 64;
  const float* er = Ee + (size_t)e * 64;

  float p0 = qr[lane]      * (kr[lane]      + er[lane]);
  float p1 = qr[lane + 32] * (kr[lane + 32] + er[lane + 32]);
  if (H == 1) { p0 += p1; p1 = 0.f; }
#pragma unroll
  for (int off = 16; off; off >>= 1) {
    p0 += __shfl_xor(p0, off, 32);
    p1 += __shfl_xor(p1, off, 32);
  }
  if (lane == 0) {
    if (H == 2) {
      const float a0 = p0 * rscale, a1 = p1 * rscale;
      alpha[(size_t)e * 2 + 0] = a0;
      alpha[(size_t)e * 2 + 1] = a1;
      atomicMax(&Menc[(size_t)dst * 2 + 0], encf(a0));
      atomicMax(&Menc[(size_t)dst * 2 + 1], encf(a1));
    } else {
      const float a0 = p0 * rscale;
      alpha[e] = a0;
      atomicMax(&Menc[dst], encf(a0));
    }
  }
}

// ---------------------------------------------------------------------------
// Pass 2: a = exp(alpha - m[dst]); Ssum[dst] += a; Acc[dst] += a*(v[src]+e)
// ---------------------------------------------------------------------------
__global__ void __launch_bounds__(256)
edge_accum_kernel(const float* __restrict__ V, const float* __restrict__ Ee,
                  const float* __restrict__ alpha, const unsigned* __restrict__ Menc,
                  const int* __restrict__ ei, float* __restrict__ Ssum,
                  float* __restrict__ Acc, int H) {
  const int e = (blockIdx.x * blockDim.x + threadIdx.x) >> 5;
  if (e >= EDGES) return;
  const int lane = threadIdx.x & 31;
  const int src = ei[e], dst = ei[EDGES + e];

  float a0, a1;
  if (H == 2) {
    a0 = __expf(alpha[(size_t)e * 2 + 0] - decf(Menc[(size_t)dst * 2 + 0]));
    a1 = __expf(alpha[(size_t)e * 2 + 1] - decf(Menc[(size_t)dst * 2 + 1]));
  } else {
    a0 = __expf(alpha[e] - decf(Menc[dst]));
    a1 = a0;
  }
  if (lane == 0) {
    atomicAdd(&Ssum[(size_t)dst * H + 0], a0);
    if (H == 2) atomicAdd(&Ssum[(size_t)dst * 2 + 1], a1);
  }
  const float* vr = V + (size_t)src * 64;
  const float* er = Ee + (size_t)e * 64;
  atomicAdd(&Acc[(size_t)dst * 64 + lane],      a0 * (vr[lane]      + er[lane]));
  atomicAdd(&Acc[(size_t)dst * 64 + lane + 32], a1 * (vr[lane + 32] + er[lane + 32]));
}

// ---------------------------------------------------------------------------
// Finalize: out = Acc/(Ssum+1e-16) + skip  (+ PReLU for layers 0,1)
// ---------------------------------------------------------------------------
__global__ void node_finalize_kernel(const float* __restrict__ Acc,
                                     const float* __restrict__ Ssum,
                                     const float* __restrict__ S,
                                     const float* __restrict__ prelu_a,
                                     float* __restrict__ Out, int n, int H,
                                     int applyPrelu) {
  const int i = blockIdx.x * blockDim.x + threadIdx.x;
  if (i >= n * 64) return;
  const int node = i >> 6, c = i & 63;
  const int h = (H == 2) ? (c >> 5) : 0;
  const float s = Ssum[(size_t)node * H + h];
  float v = Acc[i] / (s + 1e-16f) + S[i];
  if (applyPrelu) {
    const float a = prelu_a[0];
    v = (v >= 0.f) ? v : a * v;
  }
  Out[i] = v;
}

// ---------------------------------------------------------------------------
// Host orchestration
// ---------------------------------------------------------------------------
extern "C" void kernel_launch(void* const* d_in, const int* in_sizes, int n_in,
                              void* d_out, int out_size, void* d_ws, size_t ws_size,
                              hipStream_t stream) {
  const float* x     = (const float*)d_in[0];
  const float* eattr = (const float*)d_in[1];
  const int*   ei    = (const int*)d_in[2];
  const float* W[3][9];  // Wq,bq,Wk,bk,Wv,bv,We,Ws,bs per layer
  for (int l = 0; l < 3; ++l)
    for (int j = 0; j < 9; ++j) W[l][j] = (const float*)d_in[3 + l * 9 + j];
  const float* prelu_a = (const float*)d_in[30];

  float* ws = (float*)d_ws;
  size_t o = 0;
  float* hbuf0 = ws + o; o += (size_t)NODES * 64;
  float* hbuf1 = ws + o; o += (size_t)NODES * 64;
  float* Qb    = ws + o; o += (size_t)NODES * 64;
  float* Kb    = ws + o; o += (size_t)NODES * 64;
  float* Vb    = ws + o; o += (size_t)NODES * 64;
  float* Sb    = ws + o; o += (size_t)NODES * 64;
  float* Eemb  = ws + o; o += (size_t)EDGES * 64;
  float* alpha = ws + o; o += (size_t)EDGES * 2;
  unsigned* Menc = (unsigned*)(ws + o); o += (size_t)NODES * 2;
  float* Ssum  = ws + o; o += (size_t)NODES * 2;
  float* Acc   = ws + o; o += (size_t)NODES * 64;

  const dim3 blk(256);
  const int gemmNodeBlocks = (NODES / 16 * 32 + 255) / 256;  // 3125 waves
  const int gemmEdgeBlocks = (EDGES / 16 * 32 + 255) / 256;  // 50000 waves
  const int edgeBlocks     = (EDGES * 32 + 255) / 256;       // wave-per-edge
  const int nodeElemBlocks = (NODES * 64 + 255) / 256;

  const float* hin = x;
  float* houts[3] = {hbuf0, hbuf1, (float*)d_out};

  for (int l = 0; l < 3; ++l) {
    const int H = (l < 2) ? 2 : 1;
    const float rscale = (l < 2) ? 0.17677669529663687f /*1/sqrt(32)*/ : 0.125f /*1/sqrt(64)*/;

    // Dense projections (K=64): Q, K, V, skip
    gemm64_wmma_kernel<64><<<gemmNodeBlocks, blk, 0, stream>>>(hin, W[l][0], W[l][1], Qb, NODES);
    gemm64_wmma_kernel<64><<<gemmNodeBlocks, blk, 0, stream>>>(hin, W[l][2], W[l][3], Kb, NODES);
    gemm64_wmma_kernel<64><<<gemmNodeBlocks, blk, 0, stream>>>(hin, W[l][4], W[l][5], Vb, NODES);
    gemm64_wmma_kernel<64><<<gemmNodeBlocks, blk, 0, stream>>>(hin, W[l][7], W[l][8], Sb, NODES);
    // Edge embedding (K=32, no bias)
    gemm64_wmma_kernel<32><<<gemmEdgeBlocks, blk, 0, stream>>>(eattr, W[l][6], nullptr, Eemb, EDGES);

    init_state_kernel<<<nodeElemBlocks, blk, 0, stream>>>(Menc, Ssum, Acc, NODES, H);
    edge_alpha_kernel<<<edgeBlocks, blk, 0, stream>>>(Qb, Kb, Eemb, ei, alpha, Menc, H, rscale);
    edge_accum_kernel<<<edgeBlocks, blk, 0, stream>>>(Vb, Eemb, alpha, Menc, ei, Ssum, Acc, H);
    node_finalize_kernel<<<nodeElemBlocks, blk, 0, stream>>>(Acc, Ssum, Sb, prelu_a,
                                                             houts[l], NODES, H, l < 2 ? 1 : 0);
    hin = houts[l];
  }
  (void)in_sizes; (void)n_in; (void)out_size; (void)ws_size;
}